// VectorQuantizer_18219251270100
// MI455X (gfx1250) — compile-verified
//
#include <hip/hip_runtime.h>
#include <hip/hip_bf16.h>

typedef __attribute__((ext_vector_type(2))) float v2f;
typedef __attribute__((ext_vector_type(8))) float v8f;

#define K_CODES 512
#define D_DIM   8
#define S_SPAT  110592   /* 48*48*48 */
#define N_ROWS  221184   /* 2 * S_SPAT */
#define BETA_C  0.25f

// ---------------------------------------------------------------------------
// Kernel 0: per-code squared norms ||e_k||^2  (512 values)
// ---------------------------------------------------------------------------
__global__ void vq_codenorm(const float* __restrict__ ew, float* __restrict__ cn) {
    int k = blockIdx.x * blockDim.x + threadIdx.x;
    if (k < K_CODES) {
        const float* e = ew + k * D_DIM;
        float s = 0.f;
#pragma unroll
        for (int c = 0; c < D_DIM; ++c) s += e[c] * e[c];
        cn[k] = s;
    }
}

__global__ void vq_zero_acc(float* acc) {
    if (threadIdx.x == 0 && blockIdx.x == 0) *acc = 0.f;
}

// Map float -> unsigned such that unsigned order == float order (total order).
__device__ __forceinline__ unsigned f32_sortable(float f) {
    unsigned u = __float_as_uint(f);
    return u ^ ((u & 0x80000000u) ? 0xFFFFFFFFu : 0x80000000u);
}

// ---------------------------------------------------------------------------
// Kernel A: WMMA argmin. One wave32 handles 16 rows x all 512 codes.
// Distance surrogate d[i][j] = ||e_j||^2 - 2*x_i.e_j computed as
//   D = (-2*X_tile) x E_tile^T + C(||e_j||^2) using two V_WMMA_F32_16X16X4_F32.
// A-layout (16x4 f32): lane m=lane&15 holds row m; VGPR0/1 = k-pair
// (lanes 0-15 -> k0,k1 ; lanes 16-31 -> k2,k3). B mirrors with N=lane&15.
// C/D layout: lane half selects rows 0-7 / 8-15 in VGPRs 0-7, col = lane&15.
// ---------------------------------------------------------------------------
__global__ void __launch_bounds__(256)
vq_argmin_wmma(const float* __restrict__ x, const float* __restrict__ ew,
               const float* __restrict__ cn, int* __restrict__ idxArr) {
    int wave = (int)((blockIdx.x * (unsigned)blockDim.x + threadIdx.x) >> 5);
    int lane = threadIdx.x & 31;
    int n0   = wave * 16;                 // first of 16 rows for this wave
    int m    = lane & 15;
    int kh   = (lane >> 4) << 1;          // 0 (lanes 0-15) or 2 (lanes 16-31)

    int b  = n0 / S_SPAT;                 // batch index (tiles never straddle)
    int s0 = n0 - b * S_SPAT;
    const float* xb = x + (size_t)b * D_DIM * S_SPAT + s0 + m;

    // A fragments, pre-scaled by -2 (channel stride = S_SPAT)
    v2f a_lo, a_hi;
    a_lo.x = -2.f * xb[(size_t)(kh + 0) * S_SPAT];
    a_lo.y = -2.f * xb[(size_t)(kh + 1) * S_SPAT];
    a_hi.x = -2.f * xb[(size_t)(kh + 4) * S_SPAT];
    a_hi.y = -2.f * xb[(size_t)(kh + 5) * S_SPAT];

    float minv[8]; int mini[8];
#pragma unroll
    for (int v = 0; v < 8; ++v) { minv[v] = 3.0e38f; mini[v] = 0; }

    for (int t = 0; t < K_CODES / 16; ++t) {
        int col = t * 16 + m;
        const float* ep = ew + col * D_DIM + kh;   // codebook row-major [512,8]
        v2f b_lo, b_hi;
        b_lo.x = ep[0]; b_lo.y = ep[1];
        b_hi.x = ep[4]; b_hi.y = ep[5];

        float nrm = cn[col];                       // bias depends only on column
        v8f c = { nrm, nrm, nrm, nrm, nrm, nrm, nrm, nrm };
        c = __builtin_amdgcn_wmma_f32_16x16x4_f32(false, a_lo, false, b_lo,
                                                  (short)0, c, false, false);
        c = __builtin_amdgcn_wmma_f32_16x16x4_f32(false, a_hi, false, b_hi,
                                                  (short)0, c, false, false);
#pragma unroll
        for (int v = 0; v < 8; ++v) {
            float cv = c[v];
            if (cv < minv[v]) { minv[v] = cv; mini[v] = col; }  // strict < keeps lowest idx
        }
    }

    // Pack (value, index) into a sortable u64 key; unsigned min == lexicographic
    // (value, then lowest index) min. Branchless 16-lane butterfly per half-wave.
    unsigned long long key[8];
#pragma unroll
    for (int v = 0; v < 8; ++v)
        key[v] = ((unsigned long long)f32_sortable(minv[v]) << 32) | (unsigned)mini[v];

#pragma unroll
    for (int off = 8; off >= 1; off >>= 1) {
#pragma unroll
        for (int v = 0; v < 8; ++v) {
            unsigned long long ok = __shfl_xor(key[v], off, 32);
            key[v] = (ok < key[v]) ? ok : key[v];
        }
    }

    if (m == 0) {
        int rbase = n0 + (lane >> 4) * 8;   // lane 0 -> rows 0-7, lane 16 -> rows 8-15
#pragma unroll
        for (int v = 0; v < 8; ++v) idxArr[rbase + v] = (int)(unsigned)key[v];
    }
}

// ---------------------------------------------------------------------------
// Kernel B: gather quantized vectors, write STE output (value == quantized),
// accumulate sum of squared error for the loss. Output is write-once ->
// non-temporal stores to keep it out of L2.
// ---------------------------------------------------------------------------
__global__ void __launch_bounds__(256)
vq_gather_out(const float* __restrict__ x, const float* __restrict__ ew,
              const int* __restrict__ idxArr, float* __restrict__ out,
              float* __restrict__ acc) {
    int n = blockIdx.x * blockDim.x + threadIdx.x;
    float sq = 0.f;
    if (n < N_ROWS) {
        int k = idxArr[n];
        int b = n / S_SPAT;
        int s = n - b * S_SPAT;
        const float* e = ew + k * D_DIM;
        size_t base = (size_t)b * D_DIM * S_SPAT + s;
#pragma unroll
        for (int c = 0; c < D_DIM; ++c) {
            float ev = e[c];
            float xv = x[base + (size_t)c * S_SPAT];
            __builtin_nontemporal_store(ev, out + base + (size_t)c * S_SPAT);
            float d = ev - xv;
            sq += d * d;
        }
    }
#pragma unroll
    for (int off = 16; off >= 1; off >>= 1) sq += __shfl_xor(sq, off, 32);
    if ((threadIdx.x & 31) == 0) atomicAdd(acc, sq);
}

// ---------------------------------------------------------------------------
// Kernel D: finalize loss = q_latent + beta*e_latent = 1.25*clip(mse, 0, 10)
// ---------------------------------------------------------------------------
__global__ void vq_loss(const float* __restrict__ acc, float* __restrict__ loss_out) {
    float mean = *acc * (1.0f / (float)(N_ROWS * D_DIM));
    float cl = fminf(fmaxf(mean, 0.f), 10.f);
    *loss_out = (1.f + BETA_C) * cl;
}

// ---------------------------------------------------------------------------
// Kernel C: one-hot encodings [N, 512] — the 453 MB store-bound phase.
// One thread per element, fully coalesced (base offset is only 4B aligned).
// 453 MB write-once stream (2.4x the 192 MB L2) -> non-temporal stores.
// ---------------------------------------------------------------------------
__global__ void __launch_bounds__(256)
vq_encodings(const int* __restrict__ idxArr, float* __restrict__ enc) {
    long long g = (long long)blockIdx.x * blockDim.x + threadIdx.x;
    int n = (int)(g >> 9);
    int k = (int)(g & (K_CODES - 1));
    float v = (idxArr[n] == k) ? 1.0f : 0.0f;
    __builtin_nontemporal_store(v, enc + g);
}

// ---------------------------------------------------------------------------
extern "C" void kernel_launch(void* const* d_in, const int* in_sizes, int n_in,
                              void* d_out, int out_size, void* d_ws, size_t ws_size,
                              hipStream_t stream) {
    const float* x  = (const float*)d_in[0];   // (2, 8, 48, 48, 48) f32
    const float* ew = (const float*)d_in[1];   // (512, 8) f32

    // workspace layout
    float* acc = (float*)d_ws;                           // 4 B
    float* cn  = (float*)((char*)d_ws + 256);            // 512 f32
    int*   idx = (int*)((char*)d_ws + 4096);             // N ints (~864 KB)

    // d_out layout: loss (1) | out (2*8*48^3) | encodings (N*512)
    float* loss = (float*)d_out;
    float* out  = loss + 1;
    float* enc  = out + (size_t)2 * D_DIM * S_SPAT;

    vq_codenorm<<<2, 256, 0, stream>>>(ew, cn);
    vq_zero_acc<<<1, 1, 0, stream>>>(acc);

    // N/16 waves, 8 waves per 256-thread block
    int waves   = N_ROWS / 16;                // 13824
    int blocksA = waves * 32 / 256;           // 1728
    vq_argmin_wmma<<<blocksA, 256, 0, stream>>>(x, ew, cn, idx);

    vq_gather_out<<<N_ROWS / 256, 256, 0, stream>>>(x, ew, idx, out, acc);
    vq_loss<<<1, 1, 0, stream>>>(acc, loss);

    long long encElems = (long long)N_ROWS * K_CODES;    // 113,246,208
    int blocksC = (int)(encElems / 256);                 // 442,368
    vq_encodings<<<blocksC, 256, 0, stream>>>(idx, enc);
}